// Descriptor_3908420239890
// MI455X (gfx1250) — compile-verified
//
#include <hip/hip_runtime.h>

typedef __attribute__((ext_vector_type(2))) float v2f;
typedef __attribute__((ext_vector_type(8))) float v8f;

// Problem constants (match reference)
constexpr int kS     = 4;      // symbol types
constexpr int kNLAB  = 16;     // kS*kS
constexpr int kC1    = 32;     // embedding layer 1 width
constexpr int kC2    = 64;     // embedding layer 2 width
constexpr int kB     = 4;      // batch
constexpr int kN     = 512;    // atoms
constexpr float kInvRcPi  = 0.5235987755982988f;  // pi / RC (RC = 6.0)
constexpr float kRC       = 6.0f;
constexpr float kSqrtL2E  = 1.2011224087864498f;  // sqrt(log2(e)); exp(-a^2) = exp2(-(a*sqrtL2E)^2)

__launch_bounds__(256)
__global__ void descriptor_kernel(const int*   __restrict__ numbers,   // [B,N]
                                  const float* __restrict__ coords,    // [B,N,3]
                                  const float* __restrict__ nuww0,     // [16]
                                  const float* __restrict__ sigmas0,   // [16]
                                  const float* __restrict__ centres0,  // [16,32]
                                  const float* __restrict__ nuww1,     // [16]
                                  const float* __restrict__ sigmas1,   // [16]
                                  const float* __restrict__ centres1,  // [16,64]
                                  float*       __restrict__ out)       // [B,N,4096]
{
    // Per-pair packed scalars: (s1, w1 (0 if masked), sigma1*sqrtL2E, bitcast(lab*64))
    __shared__ float4 sPk[kN];
    __shared__ float4 sRaij[kN];          // [fc, ux*fc, uy*fc, uz*fc]
    __shared__ float  sC0[kNLAB * kC1];   // centres0
    __shared__ float  sC1[kNLAB * kC2];   // centres1
    __shared__ float  sP0[kNLAB], sP1[kNLAB], sP2[kNLAB], sP3[kNLAB];
    __shared__ float  sPart[4][kC2][4];   // j-quarter partial Lm
    __shared__ float  sLm[kC2][4];        // final Lm (64x4)
    __shared__ alignas(16) float sRed[8]; // per-wave Frobenius partials

    const int t   = threadIdx.x;
    const int blk = blockIdx.x;           // atom index over B*N
    const int b   = blk >> 9;             // / 512
    const int i   = blk & (kN - 1);

    // ---- stage tables into LDS ----
    for (int k = t; k < kNLAB * kC1; k += 256) sC0[k] = centres0[k];
    for (int k = t; k < kNLAB * kC2; k += 256) sC1[k] = centres1[k];
    if (t < kNLAB) {
        sP0[t] = nuww0[t];
        sP1[t] = sigmas0[t] * kSqrtL2E;   // pre-fold exp->exp2 conversion into sigma
        sP2[t] = nuww1[t];
        sP3[t] = sigmas1[t] * kSqrtL2E;
    }

    const float xi = coords[(b * kN + i) * 3 + 0];
    const float yi = coords[(b * kN + i) * 3 + 1];
    const float zi = coords[(b * kN + i) * 3 + 2];
    const int   ti = numbers[b * kN + i];

    __syncthreads();

    // ---- Phase 1: per-pair scalars (2 j per thread) ----
    for (int jj = t; jj < kN; jj += 256) {
        const float rx = coords[(b * kN + jj) * 3 + 0] - xi;  // x_j - x_i
        const float ry = coords[(b * kN + jj) * 3 + 1] - yi;
        const float rz = coords[(b * kN + jj) * 3 + 2] - zi;
        const float d2 = rx * rx + ry * ry + rz * rz;
        const bool  nz = d2 > 0.0f;
        const float safe = sqrtf(nz ? d2 : 1.0f);
        const float d    = nz ? safe : 0.0f;
        const float inv  = 1.0f / safe;
        const float fc   = 0.5f * __cosf(d * kInvRcPi) + 0.5f;
        const bool  msk  = (d <= kRC) && (jj != i);

        const int   lab = ti * kS + numbers[b * kN + jj];
        const float w0  = sP0[lab];
        const float sg0 = sP1[lab];                 // sigma0 * sqrt(log2 e)
        const float* cp = &sC0[lab * kC1];
        float acc = 0.0f;
        #pragma unroll
        for (int c = 0; c < kC1; ++c) {
            const float al = (fc - cp[c]) * sg0;
            acc += __builtin_amdgcn_exp2f(-(al * al));   // native v_exp_f32
        }
        sPk[jj] = make_float4(w0 * acc,                      // s1
                              msk ? sP2[lab] : 0.0f,         // w1 (mask folded in)
                              sP3[lab],                      // sigma1 * sqrt(log2 e)
                              __int_as_float(lab * kC2));    // centres1 row offset
        sRaij[jj] = make_float4(fc, rx * inv * fc, ry * inv * fc, rz * inv * fc);
    }
    __syncthreads();

    // ---- Phase 2: Lm[c][f] = sum_j phi2[j][c] * raij[j][f]  (1 exp per (j,c)) ----
    {
        const int c = t & 63;         // channel (within-wave j is uniform -> LDS broadcast)
        const int g = t >> 6;         // j-quarter
        float a0 = 0.f, a1 = 0.f, a2 = 0.f, a3 = 0.f;
        const int j0 = g * (kN / 4), j1 = j0 + (kN / 4);
        #pragma unroll 4
        for (int j = j0; j < j1; ++j) {
            const float4 pk = sPk[j];                       // one ds_load_b128
            const float  al = (pk.x - sC1[__float_as_int(pk.w) + c]) * pk.z;
            const float  e  = pk.y * __builtin_amdgcn_exp2f(-(al * al));
            const float4 ra = sRaij[j];                     // one ds_load_b128
            a0 += e * ra.x; a1 += e * ra.y; a2 += e * ra.z; a3 += e * ra.w;
        }
        sPart[g][c][0] = a0; sPart[g][c][1] = a1;
        sPart[g][c][2] = a2; sPart[g][c][3] = a3;
    }
    __syncthreads();
    {
        const int c = t >> 2, f = t & 3;
        sLm[c][f] = sPart[0][c][f] + sPart[1][c][f] + sPart[2][c][f] + sPart[3][c][f];
    }
    __syncthreads();

    // ---- Phase 3: R = Lm * Lm^T via V_WMMA_F32_16X16X4_F32 (K=4 exact fit) ----
    const int wave = t >> 5;
    const int lane = t & 31;
    const int row  = lane & 15;
    const int kh   = lane >> 4;     // 0: K=0,1   1: K=2,3  (32-bit 16x4 A layout)

    v8f rt[2];
    int m0s[2], n0s[2];
    float local = 0.0f;
    #pragma unroll
    for (int tt = 0; tt < 2; ++tt) {
        const int T  = wave * 2 + tt;       // 16 tiles over 8 waves
        const int m0 = (T >> 2) * 16;
        const int n0 = (T & 3) * 16;
        m0s[tt] = m0; n0s[tt] = n0;
        v2f a, bb;
        a.x  = sLm[m0 + row][kh * 2 + 0];
        a.y  = sLm[m0 + row][kh * 2 + 1];
        bb.x = sLm[n0 + row][kh * 2 + 0];   // B = Lm^T: identical load pattern
        bb.y = sLm[n0 + row][kh * 2 + 1];
        v8f cacc = {};
        cacc = __builtin_amdgcn_wmma_f32_16x16x4_f32(
            /*neg_a=*/false, a, /*neg_b=*/false, bb,
            /*c_mod=*/(short)0, cacc, /*reuse_a=*/false, /*reuse_b=*/false);
        rt[tt] = cacc;
        #pragma unroll
        for (int v = 0; v < 8; ++v) local += cacc[v] * cacc[v];
    }

    // ---- Phase 4: Frobenius norm (wave32 butterfly + 8-word LDS) + store ----
    #pragma unroll
    for (int m = 16; m >= 1; m >>= 1) local += __shfl_xor(local, m, 32);
    if (lane == 0) sRed[wave] = local;
    __syncthreads();
    const float4 r0 = *(const float4*)&sRed[0];
    const float4 r1 = *(const float4*)&sRed[4];
    const float  tot = (r0.x + r0.y + r0.z + r0.w) + (r1.x + r1.y + r1.z + r1.w);
    const float  scale = rsqrtf(tot);

    float* outp = out + (size_t)blk * (kC2 * kC2);
    #pragma unroll
    for (int tt = 0; tt < 2; ++tt) {
        #pragma unroll
        for (int v = 0; v < 8; ++v) {
            // C/D layout: VGPR v -> M = v + 8*kh, N = row
            outp[(m0s[tt] + v + 8 * kh) * kC2 + n0s[tt] + row] = rt[tt][v] * scale;
        }
    }
}

extern "C" void kernel_launch(void* const* d_in, const int* in_sizes, int n_in,
                              void* d_out, int out_size, void* d_ws, size_t ws_size,
                              hipStream_t stream) {
    // input order: boxs(0, unused), numbers(1), coords(2),
    //              nuww0(3), sigmas0(4), centres0(5), nuww1(6), sigmas1(7), centres1(8)
    const int*   numbers  = (const int*)  d_in[1];
    const float* coords   = (const float*)d_in[2];
    const float* nuww0    = (const float*)d_in[3];
    const float* sigmas0  = (const float*)d_in[4];
    const float* centres0 = (const float*)d_in[5];
    const float* nuww1    = (const float*)d_in[6];
    const float* sigmas1  = (const float*)d_in[7];
    const float* centres1 = (const float*)d_in[8];
    float* out = (float*)d_out;

    descriptor_kernel<<<dim3(kB * kN), dim3(256), 0, stream>>>(
        numbers, coords, nuww0, sigmas0, centres0, nuww1, sigmas1, centres1, out);
}